// SSIMLoss_77687368450180
// MI455X (gfx1250) — compile-verified
//
#include <hip/hip_runtime.h>

typedef float v2f __attribute__((ext_vector_type(2)));
typedef float v8f __attribute__((ext_vector_type(8)));

#define TILE   16
#define PADR   3
#define HALO   22      // TILE + 2*PADR
#define XPITCH 24      // halo pitch padded to K=24 (multiple of 4, keeps b64 alignment)
#define NCHUNK 6       // 24 / 4
#define R7     (1.0f/7.0f)
#define FIXSCALE 16777216.0f   // 2^24 fixed point for deterministic integer accumulation

// ---- monotonic uint key for float atomic max/min (exact, order independent) ----
__device__ __forceinline__ unsigned fkey(float f) {
  unsigned b = __float_as_uint(f);
  return (b & 0x80000000u) ? ~b : (b | 0x80000000u);
}
__device__ __forceinline__ float funkey(unsigned k) {
  unsigned b = (k & 0x80000000u) ? (k & 0x7FFFFFFFu) : ~k;
  return __uint_as_float(b);
}

__global__ void k_init(unsigned long long* gsum, unsigned* mmax, unsigned* mmin) {
  *gsum = 0ull;
  *mmax = 0u;           // key of -inf side
  *mmin = 0xFFFFFFFFu;  // key of +inf side
}

__global__ void k_minmax(const float* __restrict__ Y, int n,
                         unsigned* mmax, unsigned* mmin) {
  float vmax = -3.402823466e38f, vmin = 3.402823466e38f;
  for (int i = blockIdx.x * blockDim.x + threadIdx.x; i < n;
       i += gridDim.x * blockDim.x) {
    float v = Y[i];
    vmax = fmaxf(vmax, v);
    vmin = fminf(vmin, v);
  }
#pragma unroll
  for (int off = 16; off; off >>= 1) {
    vmax = fmaxf(vmax, __shfl_xor(vmax, off, 32));
    vmin = fminf(vmin, __shfl_xor(vmin, off, 32));
  }
  __shared__ float smax[8], smin[8];
  int wid = threadIdx.x >> 5;
  if ((threadIdx.x & 31) == 0) { smax[wid] = vmax; smin[wid] = vmin; }
  __syncthreads();
  if (threadIdx.x == 0) {
    int nw = blockDim.x >> 5;
    for (int w = 1; w < nw; ++w) {
      vmax = fmaxf(vmax, smax[w]);
      vmin = fminf(vmin, smin[w]);
    }
    atomicMax(mmax, fkey(vmax));
    atomicMin(mmin, fkey(vmin));
  }
}

// One wave per 16x16 output tile of one channel plane.
// Separable 7-tap box filter as banded matmuls on the WMMA pipe:
//   H_p(24x16)  = In_p(16x24) x S(24x16)   (two row-tiles cover the 22 halo rows)
//   Out_p(16x16)= S(16x24)    x H_p(24x16)
// S[k,j] = 1/7 iff 0 <= k-j <= 6 else 0; identical band for both passes,
// hoisted into 6 v2f registers and reused across all 5 maps.
__global__ __launch_bounds__(32)
void k_ssim(const float* __restrict__ X, const float* __restrict__ Y,
            const unsigned* __restrict__ mmax, const unsigned* __restrict__ mmin,
            unsigned long long* __restrict__ gsum, int tilesX, int tilesY) {
  __shared__ float Xs[HALO * XPITCH];
  __shared__ float Ys[HALO * XPITCH];
  __shared__ float Hs[5][24 * 16];

  const int lane = threadIdx.x;
  const int j  = lane & 15;   // column / M index within 16
  const int hi = lane >> 4;   // lane-half selects K pair per WMMA layout

  int t = blockIdx.x;
  const int tx = (t % tilesX) * TILE; t /= tilesX;
  const int ty = (t % tilesY) * TILE; t /= tilesY;
  const long long base = (long long)t * 512 * 512;  // t = channel plane 0..95

  // ---- cooperative halo load with zero padding (matches conv SAME zero-pad) ----
  for (int i = lane; i < HALO * HALO; i += 32) {
    int r = i / HALO, c = i % HALO;
    int gy = ty - PADR + r, gx = tx - PADR + c;
    bool ok = (gy >= 0) && (gy < 512) && (gx >= 0) && (gx < 512);
    long long gidx = base + (long long)gy * 512 + gx;
    Xs[r * XPITCH + c] = ok ? X[gidx] : 0.0f;
    Ys[r * XPITCH + c] = ok ? Y[gidx] : 0.0f;
  }
  // zero the K-padding columns 22..23 and intermediate rows 22..23 of each map
  for (int r = lane; r < HALO; r += 32) {
    Xs[r * XPITCH + 22] = 0.0f; Xs[r * XPITCH + 23] = 0.0f;
    Ys[r * XPITCH + 22] = 0.0f; Ys[r * XPITCH + 23] = 0.0f;
  }
#pragma unroll
  for (int p = 0; p < 5; ++p) Hs[p][22 * 16 + lane] = 0.0f;  // rows 22,23
  __syncthreads();

  // ---- hoisted banded constants: S chunk c, rows {4c+2hi, 4c+2hi+1}, col j ----
  v2f band[NCHUNK];
#pragma unroll
  for (int c = 0; c < NCHUNK; ++c) {
    int d0 = 4 * c + 2 * hi - j;
    band[c].x = ((unsigned)d0       <= 6u) ? R7 : 0.0f;
    band[c].y = ((unsigned)(d0 + 1) <= 6u) ? R7 : 0.0f;
  }

  // ---------- horizontal pass: load X/Y once per chunk, feed all 5 maps ----------
#pragma unroll
  for (int dt = 0; dt < 2; ++dt) {
    int mrow = dt * 16 + j;
    if (mrow > 21) mrow = 21;  // clamp: rows 22..31 of tile 1 are discarded
    v8f acc[5];
#pragma unroll
    for (int p = 0; p < 5; ++p) acc[p] = (v8f)0.0f;
#pragma unroll
    for (int c = 0; c < NCHUNK; ++c) {
      int k = 4 * c + 2 * hi;  // A layout: lanes 0-15 hold K{0,1}, 16-31 hold K{2,3}
      v2f xv = *(const v2f*)&Xs[mrow * XPITCH + k];
      v2f yv = *(const v2f*)&Ys[mrow * XPITCH + k];
      v2f av[5];
      av[0] = xv; av[1] = yv; av[2] = xv * xv; av[3] = yv * yv; av[4] = xv * yv;
      v2f bv = band[c];
#pragma unroll
      for (int p = 0; p < 5; ++p)
        acc[p] = __builtin_amdgcn_wmma_f32_16x16x4_f32(
            false, av[p], false, bv, (short)0, acc[p], false, false);
    }
    // scatter D (C/D layout: VGPR v -> row v + 8*hi) into the 5 H buffers
    if (dt == 0) {
#pragma unroll
      for (int p = 0; p < 5; ++p)
#pragma unroll
        for (int v = 0; v < 8; ++v) Hs[p][(v + 8 * hi) * 16 + j] = acc[p][v];
    } else if (hi == 0) {
#pragma unroll
      for (int p = 0; p < 5; ++p)
#pragma unroll
        for (int v = 0; v < 6; ++v) Hs[p][(16 + v) * 16 + j] = acc[p][v];
    }
  }
  __syncthreads();

  // ---------- vertical pass: constant banded A, data B from Hs ----------
  v8f res[5];
#pragma unroll
  for (int p = 0; p < 5; ++p) res[p] = (v8f)0.0f;
#pragma unroll
  for (int p = 0; p < 5; ++p) {
#pragma unroll
    for (int c = 0; c < NCHUNK; ++c) {
      int k = 4 * c + 2 * hi;  // B layout: VGPR0 row k, VGPR1 row k+1, col j
      v2f b;
      b.x = Hs[p][k * 16 + j];
      b.y = Hs[p][k * 16 + 16 + j];
      res[p] = __builtin_amdgcn_wmma_f32_16x16x4_f32(
          false, band[c], false, b, (short)0, res[p], false, false);
    }
  }

  // ---------- elementwise SSIM + deterministic reduction ----------
  const float dr  = funkey(*mmax) - funkey(*mmin);
  const float C1v = (0.01f * dr) * (0.01f * dr);
  const float C2v = (0.03f * dr) * (0.03f * dr);

  float lsum = 0.0f;
#pragma unroll
  for (int v = 0; v < 8; ++v) {
    float mu1 = res[0][v], mu2 = res[1][v];
    float exx = res[2][v], eyy = res[3][v], exy = res[4][v];
    float mu1s = mu1 * mu1, mu2s = mu2 * mu2, mu12 = mu1 * mu2;
    float s1 = exx - mu1s, s2 = eyy - mu2s, s12 = exy - mu12;
    float num = (2.0f * mu12 + C1v) * (2.0f * s12 + C2v);
    float den = (mu1s + mu2s + C1v) * (s1 + s2 + C2v);
    lsum += num * __builtin_amdgcn_rcpf(den);  // ~1 ulp; fine for a 25M-sample mean
  }
#pragma unroll
  for (int off = 16; off; off >>= 1) lsum += __shfl_xor(lsum, off, 32);
  if (lane == 0) {
    long long q = (long long)(lsum * FIXSCALE);  // fixed-point: deterministic atomics
    atomicAdd(gsum, (unsigned long long)q);
  }
}

__global__ void k_final(const unsigned long long* __restrict__ gsum,
                        float* __restrict__ out, double n) {
  long long s = (long long)*gsum;
  out[0] = 1.0f - (float)((double)s / (double)FIXSCALE / n);
}

extern "C" void kernel_launch(void* const* d_in, const int* in_sizes, int n_in,
                              void* d_out, int out_size, void* d_ws, size_t ws_size,
                              hipStream_t stream) {
  const float* X = (const float*)d_in[0];
  const float* Y = (const float*)d_in[1];
  float* out = (float*)d_out;

  unsigned long long* gsum = (unsigned long long*)d_ws;
  unsigned* mmax = (unsigned*)((char*)d_ws + 8);
  unsigned* mmin = (unsigned*)((char*)d_ws + 12);

  const int nY = in_sizes[1];

  k_init<<<1, 1, 0, stream>>>(gsum, mmax, mmin);
  k_minmax<<<1024, 256, 0, stream>>>(Y, nY, mmax, mmin);

  const int tilesX = 512 / TILE, tilesY = 512 / TILE;
  const int nch = 32 * 3;
  const int nblocks = tilesX * tilesY * nch;  // 98,304 tiles, one wave each
  k_ssim<<<nblocks, 32, 0, stream>>>(X, Y, mmax, mmin, gsum, tilesX, tilesY);

  k_final<<<1, 1, 0, stream>>>(gsum, out, 25165824.0 /* 32*3*512*512 */);
}